// GumbelTopK_68994354643627
// MI455X (gfx1250) — compile-verified
//
#include <hip/hip_runtime.h>
#include <cstdint>
#include <cstddef>

// Gumbel soft-top-k, one row (4096 f32) per 256-thread workgroup.
// Row stays register-resident across all k softmax iterations; HBM is
// touched exactly once per element (read) plus once for the output.
// Staging uses CDNA5 async global->LDS DMA (ASYNCcnt).

#define ROW_N       4096
#define THREADS     256
#define F4_CHUNKS   4              // (ROW_N/4) float4 slots / THREADS
#define NWAVES      (THREADS / 32)

#define TEMPERATURE     1.0f
#define MIN_TEMPERATURE 0.01f
#define PENALTY         1.0e6f

__device__ __forceinline__ unsigned lds_lo32(const void* p) {
  // Generic pointers into the LDS aperture carry the workgroup-relative
  // LDS byte offset in their low 32 bits.
  return (unsigned)(uintptr_t)p;
}

__global__ __launch_bounds__(THREADS)
void GumbelTopK_68994354643627_kernel(const float* __restrict__ logits,
                                      const float* __restrict__ gumbel,
                                      const int*   __restrict__ kptr,
                                      float*       __restrict__ out)
{
  __shared__ float4 ldsA[ROW_N / 4];   // 16 KB: logits row
  __shared__ float4 ldsB[ROW_N / 4];   // 16 KB: gumbel row
  __shared__ float  redMax[NWAVES];
  __shared__ float  redSum[NWAVES];

  const int row  = blockIdx.x;
  const int tid  = threadIdx.x;
  const int wid  = tid >> 5;
  const int lane = tid & 31;

  const float4* __restrict__ gA = (const float4*)(logits + (size_t)row * ROW_N);
  const float4* __restrict__ gB = (const float4*)(gumbel + (size_t)row * ROW_N);

  // ---- Stage row into LDS with async DMA (CDNA5 ASYNCcnt path) ----
  #pragma unroll
  for (int c = 0; c < F4_CHUNKS; ++c) {
    const int idx = c * THREADS + tid;
    unsigned la = lds_lo32(&ldsA[idx]);
    unsigned lb = lds_lo32(&ldsB[idx]);
    unsigned long long ga = (unsigned long long)(uintptr_t)(gA + idx);
    unsigned long long gb = (unsigned long long)(uintptr_t)(gB + idx);
    asm volatile("global_load_async_to_lds_b128 %0, %1, off"
                 :: "v"(la), "v"(ga) : "memory");
    asm volatile("global_load_async_to_lds_b128 %0, %1, off"
                 :: "v"(lb), "v"(gb) : "memory");
  }
  // Each lane only reads back slots whose loads it issued itself, so its
  // own ASYNCcnt drain is a sufficient ordering guarantee (no barrier).
  asm volatile("s_wait_asynccnt 0" ::: "memory");

  const float invT = 1.0f / fmaxf(TEMPERATURE, MIN_TEMPERATURE);

  float4 v[F4_CHUNKS];     // "remaining" logits, register resident
  float4 msk[F4_CHUNKS];   // accumulated soft mask
  #pragma unroll
  for (int c = 0; c < F4_CHUNKS; ++c) {
    const int idx = c * THREADS + tid;
    float4 a = ldsA[idx];
    float4 b = ldsB[idx];
    v[c].x = (a.x + b.x) * invT;
    v[c].y = (a.y + b.y) * invT;
    v[c].z = (a.z + b.z) * invT;
    v[c].w = (a.w + b.w) * invT;
    msk[c] = make_float4(0.f, 0.f, 0.f, 0.f);
  }

  const int k = kptr[0];

  for (int it = 0; it < k; ++it) {
    // ---- row max (stable softmax) ----
    float mx = -3.402823466e38f;
    #pragma unroll
    for (int c = 0; c < F4_CHUNKS; ++c) {
      mx = fmaxf(mx, fmaxf(fmaxf(v[c].x, v[c].y), fmaxf(v[c].z, v[c].w)));
    }
    #pragma unroll
    for (int off = 16; off >= 1; off >>= 1)
      mx = fmaxf(mx, __shfl_xor(mx, off, 32));
    if (lane == 0) redMax[wid] = mx;
    __syncthreads();
    float rowMax = redMax[0];
    #pragma unroll
    for (int w = 1; w < NWAVES; ++w) rowMax = fmaxf(rowMax, redMax[w]);

    // ---- exp + row sum ----
    float4 e[F4_CHUNKS];
    float s = 0.f;
    #pragma unroll
    for (int c = 0; c < F4_CHUNKS; ++c) {
      e[c].x = __expf(v[c].x - rowMax);
      e[c].y = __expf(v[c].y - rowMax);
      e[c].z = __expf(v[c].z - rowMax);
      e[c].w = __expf(v[c].w - rowMax);
      s += (e[c].x + e[c].y) + (e[c].z + e[c].w);
    }
    #pragma unroll
    for (int off = 16; off >= 1; off >>= 1)
      s += __shfl_xor(s, off, 32);
    if (lane == 0) redSum[wid] = s;
    __syncthreads();
    float tot = 0.f;
    #pragma unroll
    for (int w = 0; w < NWAVES; ++w) tot += redSum[w];
    const float inv = 1.0f / tot;

    // ---- accumulate mask, suppress selected mass ----
    #pragma unroll
    for (int c = 0; c < F4_CHUNKS; ++c) {
      float px = e[c].x * inv, py = e[c].y * inv,
            pz = e[c].z * inv, pw = e[c].w * inv;
      msk[c].x += px; msk[c].y += py; msk[c].z += pz; msk[c].w += pw;
      v[c].x -= px * PENALTY; v[c].y -= py * PENALTY;
      v[c].z -= pz * PENALTY; v[c].w -= pw * PENALTY;
    }
  }

  // ---- clip to [0,1] and store ----
  float4* __restrict__ gO = (float4*)(out + (size_t)row * ROW_N);
  #pragma unroll
  for (int c = 0; c < F4_CHUNKS; ++c) {
    const int idx = c * THREADS + tid;
    float4 r;
    r.x = fminf(fmaxf(msk[c].x, 0.f), 1.f);
    r.y = fminf(fmaxf(msk[c].y, 0.f), 1.f);
    r.z = fminf(fmaxf(msk[c].z, 0.f), 1.f);
    r.w = fminf(fmaxf(msk[c].w, 0.f), 1.f);
    gO[idx] = r;
  }
}

extern "C" void kernel_launch(void* const* d_in, const int* in_sizes, int n_in,
                              void* d_out, int out_size, void* d_ws, size_t ws_size,
                              hipStream_t stream) {
  (void)n_in; (void)d_ws; (void)ws_size; (void)out_size;
  const float* logits = (const float*)d_in[0];
  const float* gumbel = (const float*)d_in[1];
  const int*   kptr   = (const int*)d_in[2];
  float*       out    = (float*)d_out;

  const int batch = in_sizes[0] / ROW_N;   // 8192 rows of 4096
  dim3 grid(batch), block(THREADS);
  GumbelTopK_68994354643627_kernel<<<grid, block, 0, stream>>>(logits, gumbel, kptr, out);
}